// GIN_53060025975177
// MI455X (gfx1250) — compile-verified
//
#include <hip/hip_runtime.h>
#include <hip/hip_bf16.h>

// ---------------------------------------------------------------------------
// GIN forward for MI455X (gfx1250).
//  - fp32 WMMA (v_wmma_f32_16x16x4_f32) for the node GEMMs (exact fp32 math).
//  - A-tile + pair-interleaved W staged in LDS: inner loop is pure
//    ds_load_b64 + v_wmma, no global traffic, no exec-mask branches.
//  - L2-resident scatter-add (all big buffers ~77MB fit in 192MB L2).
// ---------------------------------------------------------------------------

#define N_NODES   50000
#define N_EDGES   800000
#define DH        128
#define N_GRAPHS  64
#define BN_EPS    1e-5f
#define ALD       132   // padded A-tile row stride (floats): bank-conflict-free

typedef float v2f __attribute__((ext_vector_type(2)));
typedef float v8f __attribute__((ext_vector_type(8)));

__device__ __forceinline__ void atomAddF(float* p, float v) {
    unsafeAtomicAdd(p, v);  // -> global_atomic_add_f32
}

// ------------------------------ utility ------------------------------------
__global__ void zero_kernel(float* __restrict__ p, int n) {
    for (int i = blockIdx.x * blockDim.x + threadIdx.x; i < n;
         i += gridDim.x * blockDim.x)
        p[i] = 0.0f;
}

__global__ void count_kernel(const int* __restrict__ batch,
                             float* __restrict__ cnt, int n) {
    int i = blockIdx.x * blockDim.x + threadIdx.x;
    if (i < n) atomAddF(&cnt[batch[i]], 1.0f);
}

// ------------------------- edge scatter-add --------------------------------
// One lane handles 4 consecutive columns of one edge (wave == one edge).
__global__ void scatter_kernel(const float* __restrict__ h,
                               const int* __restrict__ ei,
                               float* __restrict__ agg, int n_edges) {
    int idx = blockIdx.x * blockDim.x + threadIdx.x;
    int e = idx >> 5;
    int l = idx & 31;
    if (e >= n_edges) return;
    int src = ei[e];
    int dst = ei[n_edges + e];
    const float4 v = *(const float4*)(h + (size_t)src * DH + l * 4);
    float* p = agg + (size_t)dst * DH + l * 4;
    atomAddF(p + 0, v.x);
    atomAddF(p + 1, v.y);
    atomAddF(p + 2, v.z);
    atomAddF(p + 3, v.w);
}

// --------------------------- WMMA fp32 GEMM --------------------------------
// out[n_rows x 128] = act( (A (+ A2)) @ W + bias )
// Block: 256 threads = 8 waves; block owns a 128-row M tile; each wave one
// 16-row slice, full 128 output columns (8 accumulator tiles).
__global__ __launch_bounds__(256) void wmma_gemm_kernel(
    const float* __restrict__ A, const float* __restrict__ A2,
    const float* __restrict__ W, const float* __restrict__ bias,
    float* __restrict__ out, int n_rows, int do_relu) {
    // W pair-interleaved: Wp[((k>>1)*128 + n)*2 + (k&1)] = W[k][n]
    __shared__ float Wp[DH * DH];      // 64 KB
    __shared__ float Al[128 * ALD];    // 67.6 KB, padded rows

    int t = threadIdx.x;
    int rowBlk = blockIdx.x * 128;

    // ---- stage W (pair-interleave K so B fragments are one ds_load_b64) ----
    for (int i = t; i < DH * DH / 2; i += 256) {
        int kp = i >> 7;     // k/2
        int n  = i & 127;
        Wp[i * 2 + 0] = W[(2 * kp + 0) * DH + n];
        Wp[i * 2 + 1] = W[(2 * kp + 1) * DH + n];
    }
    // ---- stage A tile, fusing x+agg; zero-pad OOB rows ----
    for (int i = t; i < 128 * DH / 4; i += 256) {
        int r  = i >> 5;
        int c4 = (i & 31) * 4;
        int gr = rowBlk + r;
        float4 v = make_float4(0.f, 0.f, 0.f, 0.f);
        if (gr < n_rows) {
            v = *(const float4*)(A + (size_t)gr * DH + c4);
            if (A2) {
                float4 w2 = *(const float4*)(A2 + (size_t)gr * DH + c4);
                v.x += w2.x; v.y += w2.y; v.z += w2.z; v.w += w2.w;
            }
        }
        float* dst = &Al[r * ALD + c4];
        dst[0] = v.x; dst[1] = v.y; dst[2] = v.z; dst[3] = v.w;
    }
    __syncthreads();

    int lane = t & 31;
    int wave = t >> 5;
    int lo = lane & 15;   // N-within-tile / M-row for A fragment
    int hi = lane >> 4;   // half-wave select
    int aBase = (wave * 16 + lo) * ALD;
    const v2f* __restrict__ Wv = (const v2f*)Wp;

    v8f acc[8];
    const v8f vzero = {0.f, 0.f, 0.f, 0.f, 0.f, 0.f, 0.f, 0.f};
#pragma unroll
    for (int i = 0; i < 8; ++i) acc[i] = vzero;

    for (int kk = 0; kk < DH / 4; ++kk) {
        // A frag (16x4): VGPR0 = K{0|2}, VGPR1 = K{1|3}; M = lane%16
        v2f a = *(const v2f*)&Al[aBase + kk * 4 + hi * 2];
        int bRow = (2 * kk + hi) * DH;
#pragma unroll
        for (int n0 = 0; n0 < 8; ++n0) {
            // B frag (4x16): VGPR r holds rows K=r (lanes0-15), K=r+2
            v2f b = Wv[bRow + n0 * 16 + lo];
            acc[n0] = __builtin_amdgcn_wmma_f32_16x16x4_f32(
                false, a, false, b, (short)0, acc[n0], false, false);
        }
    }

    // Epilogue: C/D layout -> VGPR r: M = rowBase + r + 8*hi, N = n0*16 + lo
    int rowBase = rowBlk + wave * 16;
#pragma unroll
    for (int n0 = 0; n0 < 8; ++n0) {
        int col = n0 * 16 + lo;
        float bv = bias[col];
#pragma unroll
        for (int r = 0; r < 8; ++r) {
            int mr = rowBase + r + hi * 8;
            if (mr < n_rows) {
                float v = acc[n0][r] + bv;
                if (do_relu) v = fmaxf(v, 0.f);
                out[(size_t)mr * DH + col] = v;
            }
        }
    }
}

// --------------------------- BatchNorm pieces ------------------------------
__global__ __launch_bounds__(256) void bn_stats_kernel(
    const float* __restrict__ t, float* __restrict__ sum,
    float* __restrict__ sumsq, int n_rows) {
    int c = threadIdx.x & (DH - 1);
    int row = blockIdx.x * 2 + (threadIdx.x >> 7);
    int stride = gridDim.x * 2;
    float s = 0.f, ss = 0.f;
    for (int r = row; r < n_rows; r += stride) {
        float v = t[(size_t)r * DH + c];
        s += v;
        ss += v * v;
    }
    atomAddF(&sum[c], s);
    atomAddF(&sumsq[c], ss);
}

__global__ void bn_finalize_kernel(const float* __restrict__ sum,
                                   const float* __restrict__ sq,
                                   const float* __restrict__ g,
                                   const float* __restrict__ beta,
                                   float* __restrict__ scsh, float invN) {
    int c = threadIdx.x;
    if (c >= DH) return;
    float mval = sum[c] * invN;
    float var = sq[c] * invN - mval * mval;
    float sc = g[c] / sqrtf(var + BN_EPS);
    scsh[c] = sc;
    scsh[DH + c] = beta[c] - mval * sc;
}

// normalize + relu + write next H + accumulate per-graph pooled sums
__global__ void bn_apply_pool_kernel(const float* __restrict__ t,
                                     const float* __restrict__ scsh,
                                     const int* __restrict__ batch,
                                     float* __restrict__ hout,
                                     float* __restrict__ pooled, int n_rows) {
    int idx = blockIdx.x * blockDim.x + threadIdx.x;
    if (idx >= n_rows * DH) return;
    int row = idx >> 7;
    int c = idx & (DH - 1);
    float v = t[idx] * scsh[c] + scsh[DH + c];
    v = fmaxf(v, 0.f);
    hout[idx] = v;
    atomAddF(&pooled[batch[row] * DH + c], v);
}

// ------------------------ prediction MLP: fc1 ------------------------------
// grid = 64 blocks (one graph each), 128 threads (one output column each)
__global__ __launch_bounds__(128) void fc1_kernel(
    const float* __restrict__ pooled,  // [3][64][128]
    const float* __restrict__ cnt,     // [64]
    const float* __restrict__ W1,      // [384][128]
    const float* __restrict__ b1, float* __restrict__ Z) {
    int gi = blockIdx.x;
    int j = threadIdx.x;
    float invc = 1.0f / fmaxf(cnt[gi], 1.0f);
    float acc = b1[j];
    for (int l = 0; l < 3; ++l) {
        const float* pl = pooled + l * N_GRAPHS * DH + gi * DH;
        const float* wl = W1 + l * DH * DH;
        for (int c = 0; c < DH; ++c)
            acc += pl[c] * invc * wl[c * DH + j];  // coalesced across threads
    }
    Z[gi * DH + j] = fmaxf(acc, 0.f);
}

// ---------------- prediction MLP: BN over graphs + fc2 ---------------------
__global__ __launch_bounds__(128) void final_bn_fc2_kernel(
    const float* __restrict__ Z, const float* __restrict__ g,
    const float* __restrict__ beta, const float* __restrict__ W2,  // [128]
    const float* __restrict__ b2, float* __restrict__ out) {
    __shared__ float Zs[N_GRAPHS * DH];
    __shared__ float SC[DH], SH[DH];
    int t = threadIdx.x;
    for (int i = t; i < N_GRAPHS * DH; i += 128) Zs[i] = Z[i];
    __syncthreads();

    {  // per-column stats over 64 graphs
        float s = 0.f, ss = 0.f;
        for (int gi = 0; gi < N_GRAPHS; ++gi) {
            float v = Zs[gi * DH + t];
            s += v;
            ss += v * v;
        }
        float mval = s / (float)N_GRAPHS;
        float var = ss / (float)N_GRAPHS - mval * mval;
        float sc = g[t] / sqrtf(var + BN_EPS);
        SC[t] = sc;
        SH[t] = beta[t] - mval * sc;
    }
    __syncthreads();

    if (t < N_GRAPHS) {
        float acc = b2[0];
        for (int j = 0; j < DH; ++j)
            acc += (Zs[t * DH + j] * SC[j] + SH[j]) * W2[j];
        out[t] = acc;
    }
}

// ------------------------------ launcher -----------------------------------
extern "C" void kernel_launch(void* const* d_in, const int* in_sizes, int n_in,
                              void* d_out, int out_size, void* d_ws,
                              size_t ws_size, hipStream_t stream) {
    const float* x      = (const float*)d_in[0];
    const int*   ei     = (const int*)d_in[1];
    const int*   batch  = (const int*)d_in[2];
    const float* lW1[3], *lb1[3], *lW2[3], *lb2[3], *lg[3], *lbeta[3];
    int p = 3;
    for (int l = 0; l < 3; ++l) {
        lW1[l]   = (const float*)d_in[p++];
        lb1[l]   = (const float*)d_in[p++];
        lW2[l]   = (const float*)d_in[p++];
        lb2[l]   = (const float*)d_in[p++];
        lg[l]    = (const float*)d_in[p++];
        lbeta[l] = (const float*)d_in[p++];
    }
    const float* pW1   = (const float*)d_in[p++];
    const float* pb1   = (const float*)d_in[p++];
    const float* pg    = (const float*)d_in[p++];
    const float* pbeta = (const float*)d_in[p++];
    const float* pW2   = (const float*)d_in[p++];
    const float* pb2   = (const float*)d_in[p++];

    const int NE = N_NODES * DH;  // 6.4M
    float* ws     = (float*)d_ws;
    float* H      = ws;            // [50000][128]
    float* T1     = H + NE;        // [50000][128]
    float* T2     = T1 + NE;       // [50000][128]  (also agg buffer)
    float* sumb   = T2 + NE;       // [128]
    float* sqb    = sumb + DH;     // [128]
    float* scsh   = sqb + DH;      // [256]
    float* pooled = scsh + 2 * DH; // [3][64][128]
    float* cntb   = pooled + 3 * N_GRAPHS * DH;  // [64]
    float* Zbuf   = cntb + N_GRAPHS;             // [64][128]

    // zero pooled sums + counts (contiguous), then count nodes/graph
    zero_kernel<<<64, 256, 0, stream>>>(pooled, 3 * N_GRAPHS * DH + N_GRAPHS);
    count_kernel<<<(N_NODES + 255) / 256, 256, 0, stream>>>(batch, cntb,
                                                            N_NODES);

    const int gemmBlocks = (N_NODES + 127) / 128;  // 391
    const float* h = x;
    for (int l = 0; l < 3; ++l) {
        // agg = scatter_add(h[src] -> dst)
        zero_kernel<<<2048, 256, 0, stream>>>(T2, NE);
        zero_kernel<<<1, 256, 0, stream>>>(sumb, 2 * DH);
        scatter_kernel<<<(N_EDGES * 32) / 256, 256, 0, stream>>>(h, ei, T2,
                                                                 N_EDGES);
        // T1 = relu((h + agg) @ W1 + b1)
        wmma_gemm_kernel<<<gemmBlocks, 256, 0, stream>>>(h, T2, lW1[l], lb1[l],
                                                         T1, N_NODES, 1);
        // T2 = T1 @ W2 + b2
        wmma_gemm_kernel<<<gemmBlocks, 256, 0, stream>>>(
            T1, nullptr, lW2[l], lb2[l], T2, N_NODES, 0);
        // BatchNorm stats -> scale/shift
        bn_stats_kernel<<<256, 256, 0, stream>>>(T2, sumb, sqb, N_NODES);
        bn_finalize_kernel<<<1, 128, 0, stream>>>(sumb, sqb, lg[l], lbeta[l],
                                                  scsh, 1.0f / (float)N_NODES);
        // H = relu(bn(T2)); pooled[l] += per-graph sums
        bn_apply_pool_kernel<<<(NE + 255) / 256, 256, 0, stream>>>(
            T2, scsh, batch, H, pooled + l * N_GRAPHS * DH, N_NODES);
        h = H;
    }

    fc1_kernel<<<N_GRAPHS, 128, 0, stream>>>(pooled, cntb, pW1, pb1, Zbuf);
    final_bn_fc2_kernel<<<1, 128, 0, stream>>>(Zbuf, pg, pbeta, pW2, pb2,
                                               (float*)d_out);
}